// SelfAttentionLayer_82102594830569
// MI455X (gfx1250) — compile-verified
//
#include <hip/hip_runtime.h>
#include <math.h>

// ---------------------------------------------------------------------------
// SelfAttentionLayer on MI455X (gfx1250, wave32).
//
//  q = x Wq^T ; k = x Wk^T ; v = x Wv^T + bv           (f32 WMMA GEMMs)
//  attn[b,i]  = sum_j softmax_j(q[b,i]*k[b,j]/32) v[b,j]  (rank-1 logits =>
//               per-(b,i) scalar c; native v_exp_f32 via exp2f)
//  out = attn Wo^T + bo                                 (f32 WMMA GEMM)
// ---------------------------------------------------------------------------

typedef __attribute__((ext_vector_type(2))) float v2f;
typedef __attribute__((ext_vector_type(8))) float v8f;

#define HDIM 1024
#define BDIM 512
#define ODIM 512

// ---------------------------------------------------------------------------
// D[M,N] = A[M,K=1024] * W[N,K=1024]^T (+ bias[n])   (all row-major f32)
//
// 256 threads = 8 wave32. Block tile 32(M) x 64(N); wave w owns the 16x16
// tile (tm = w&1, tn = w>>1). K staged through LDS 32 at a time; WMMA
// f32 16x16x4 accumulates in f32 registers (v8f).
//
// Fragment layouts (CDNA5 ISA 7.12.2):
//   A 16x4 f32 : lane half h (lane>>4), VGPR j -> K = 2h + j, M = lane&15
//   B 4x16 f32 : assumed mirrored K mapping, N = lane&15 (compile-only env;
//                documented assumption)
//   C/D 16x16  : VGPR r -> M = r + 8*h, N = lane&15
// ---------------------------------------------------------------------------
__global__ __launch_bounds__(256)
void gemm_nt_wmma_f32(const float* __restrict__ A, const float* __restrict__ W,
                      const float* __restrict__ bias, float* __restrict__ D,
                      int N)
{
    constexpr int K = HDIM;
    constexpr int LDP = 36;                 // padded LDS row (36*4 = 144B, 16B aligned)
    __shared__ float sA[32 * LDP];
    __shared__ float sW[64 * LDP];

    const int t    = threadIdx.x;
    const int lane = t & 31;                // wave32
    const int wave = t >> 5;                // 0..7
    const int tm   = wave & 1;              // 0..1 : 16-row tile in block
    const int tn   = wave >> 1;             // 0..3 : 16-col tile in block
    const int m0   = blockIdx.y * 32;
    const int n0   = blockIdx.x * 64;
    const int h    = lane >> 4;             // lane half
    const int ln   = lane & 15;
    const int kh   = 2 * h;

    // cooperative loads: thread t -> row t>>3, 4 consecutive K elements
    const int ar = t >> 3;                  // 0..31
    const int ac = (t & 7) << 2;            // 0,4,...,28
    const float* Aload  = A + (size_t)(m0 + ar) * K + ac;
    const float* Wload0 = W + (size_t)(n0 + ar) * K + ac;
    const float* Wload1 = W + (size_t)(n0 + 32 + ar) * K + ac;

    const int arow = (tm * 16 + ln) * LDP;
    const int brow = (tn * 16 + ln) * LDP;

    v8f acc{};
    for (int k0 = 0; k0 < K; k0 += 32) {
        float4 av = *(const float4*)(Aload  + k0);
        float4 w0 = *(const float4*)(Wload0 + k0);
        float4 w1 = *(const float4*)(Wload1 + k0);
        *(float4*)&sA[ar * LDP + ac]        = av;
        *(float4*)&sW[ar * LDP + ac]        = w0;
        *(float4*)&sW[(ar + 32) * LDP + ac] = w1;
        __syncthreads();

#pragma unroll
        for (int kk = 0; kk < 32; kk += 4) {
            v2f a, b;
            a[0] = sA[arow + kk + kh];
            a[1] = sA[arow + kk + kh + 1];
            b[0] = sW[brow + kk + kh];
            b[1] = sW[brow + kk + kh + 1];
            // 8-arg form: (neg_a, A, neg_b, B, c_mod, C, reuse_a, reuse_b)
            acc = __builtin_amdgcn_wmma_f32_16x16x4_f32(
                false, a, false, b, (short)0, acc, false, false);
        }
        __syncthreads();
    }

    const int   col = n0 + tn * 16 + ln;
    const float bb  = bias ? bias[col] : 0.0f;
    const int rbase = m0 + tm * 16 + (h ? 8 : 0);
#pragma unroll
    for (int r = 0; r < 8; ++r)
        D[(size_t)(rbase + r) * N + col] = acc[r] + bb;
}

// ---------------------------------------------------------------------------
// attn[b,i] = sum_j exp2(c*k_j - m) * v_j / sum_j exp2(c*k_j - m)
//   with c = q[b,i] * log2(e)/32, m = max(c*kmax, c*kmin) (sign-safe max).
// One block per batch row b: k[b,:], v[b,:] live in LDS (8KB), each of the
// 256 threads produces 4 outputs (4 x 1024 native v_exp_f32 per thread).
// ---------------------------------------------------------------------------
__global__ __launch_bounds__(256)
void attn_softmax_rank1(const float* __restrict__ Q, const float* __restrict__ Km,
                        const float* __restrict__ V, float* __restrict__ O)
{
    __shared__ float sk[HDIM];
    __shared__ float sv[HDIM];
    __shared__ float red[256];
    __shared__ float s_kmax, s_kmin;

    const int b = blockIdx.x;
    const int t = threadIdx.x;

    const float4* kg = (const float4*)(Km + (size_t)b * HDIM);
    const float4* vg = (const float4*)(V  + (size_t)b * HDIM);
    float4 ka = kg[t];                       // elements 4t..4t+3
    ((float4*)sk)[t] = ka;
    ((float4*)sv)[t] = vg[t];

    float lmax = fmaxf(fmaxf(ka.x, ka.y), fmaxf(ka.z, ka.w));
    float lmin = fminf(fminf(ka.x, ka.y), fminf(ka.z, ka.w));

    red[t] = lmax; __syncthreads();
    for (int s = 128; s > 0; s >>= 1) {
        if (t < s) red[t] = fmaxf(red[t], red[t + s]);
        __syncthreads();
    }
    if (t == 0) s_kmax = red[0];
    __syncthreads();
    red[t] = lmin; __syncthreads();
    for (int s = 128; s > 0; s >>= 1) {
        if (t < s) red[t] = fminf(red[t], red[t + s]);
        __syncthreads();
    }
    if (t == 0) s_kmin = red[0];
    __syncthreads();

    const float kmax = s_kmax, kmin = s_kmin;
    // scale = (1/sqrt(H)) * log2(e): work entirely in the exp2 domain so the
    // inner loop is pure v_fma + native v_exp_f32.
    const float cscale = 0.03125f * 1.4426950408889634f;
    const float4* sk4 = (const float4*)sk;
    const float4* sv4 = (const float4*)sv;

#pragma unroll
    for (int ii = 0; ii < 4; ++ii) {
        const int i = t + ii * 256;
        const float c = Q[(size_t)b * HDIM + i] * cscale;
        const float m = fmaxf(c * kmax, c * kmin);   // correct for either sign of c
        float num = 0.0f, den = 0.0f;
#pragma unroll 4
        for (int j = 0; j < HDIM / 4; ++j) {
            float4 kk = sk4[j];
            float4 vv = sv4[j];
            float e0 = exp2f(fmaf(c, kk.x, -m));
            float e1 = exp2f(fmaf(c, kk.y, -m));
            float e2 = exp2f(fmaf(c, kk.z, -m));
            float e3 = exp2f(fmaf(c, kk.w, -m));
            den += (e0 + e1) + (e2 + e3);
            num = fmaf(e0, vv.x, num);
            num = fmaf(e1, vv.y, num);
            num = fmaf(e2, vv.z, num);
            num = fmaf(e3, vv.w, num);
        }
        O[(size_t)b * HDIM + i] = num / den;
    }
}

// ---------------------------------------------------------------------------
extern "C" void kernel_launch(void* const* d_in, const int* in_sizes, int n_in,
                              void* d_out, int out_size, void* d_ws, size_t ws_size,
                              hipStream_t stream)
{
    (void)in_sizes; (void)n_in; (void)out_size; (void)ws_size;
    const float* x  = (const float*)d_in[0];
    const float* Wq = (const float*)d_in[1];
    const float* Wk = (const float*)d_in[2];
    const float* Wv = (const float*)d_in[3];
    const float* bv = (const float*)d_in[4];
    const float* Wo = (const float*)d_in[5];
    const float* bo = (const float*)d_in[6];
    float* out = (float*)d_out;

    // workspace: Q | K | V | attn  (each B*H f32 = 2MB, total 8MB)
    float* ws = (float*)d_ws;
    float* Qb = ws;
    float* Kb = ws + (size_t)BDIM * HDIM;
    float* Vb = ws + 2 * (size_t)BDIM * HDIM;
    float* At = ws + 3 * (size_t)BDIM * HDIM;

    dim3 blk(256);
    dim3 gQKV(HDIM / 64, BDIM / 32);   // (16,16): 256 blocks
    gemm_nt_wmma_f32<<<gQKV, blk, 0, stream>>>(x, Wq, nullptr, Qb, HDIM);
    gemm_nt_wmma_f32<<<gQKV, blk, 0, stream>>>(x, Wk, nullptr, Kb, HDIM);
    gemm_nt_wmma_f32<<<gQKV, blk, 0, stream>>>(x, Wv, bv,      Vb, HDIM);

    attn_softmax_rank1<<<dim3(BDIM), blk, 0, stream>>>(Qb, Kb, Vb, At);

    dim3 gO(ODIM / 64, BDIM / 32);     // (8,16)
    gemm_nt_wmma_f32<<<gO, blk, 0, stream>>>(At, Wo, bo, out, ODIM);
}